// Forward_projection_73598559584914
// MI455X (gfx1250) — compile-verified
//
#include <hip/hip_runtime.h>

#define N_IMG 128
#define PITCH 129          // +1 pad: LDS bank = (row + col) & 63, breaks column serialization
#define N_ANG 285
#define N_DET 183
#define NGRP  12           // ceil(183/16) detector groups of 16
#define NBLK  90           // blocks per batch image

typedef __attribute__((ext_vector_type(2))) float v2f;
typedef __attribute__((ext_vector_type(8))) float v8f;

__global__ __launch_bounds__(256) void radon_fwd_kernel(const float* __restrict__ x,
                                                        float* __restrict__ out) {
  extern __shared__ float simg[];   // 128 x 129 f32 = 66048 B (LDS is 320KB/WGP)

  const double RHO   = 28.284271247461902909;   // 20*sqrt(2)
  const double DTd   = 2.0 * RHO / 384.0;
  const double SSTEP = 2.0 * RHO / 183.0;
  const double IDX   = 1.0 / 0.3125;            // 1/DX

  const int batch = blockIdx.x / NBLK;
  const int blk   = blockIdx.x % NBLK;
  const float* img = x + (size_t)batch * (N_IMG * N_IMG);

  // ---- Stage image into LDS via CDNA5 async global->LDS (ASYNCcnt path) ----
  for (int q = threadIdx.x; q < N_IMG * N_IMG; q += 256) {
    int row = q >> 7, col = q & 127;
    unsigned loff = (unsigned)(unsigned long long)(&simg[row * PITCH + col]);
    unsigned long long g = (unsigned long long)(const void*)(img + q);
    asm volatile("global_load_async_to_lds_b32 %0, %1, off"
                 :: "v"(loff), "v"(g) : "memory");
  }
  asm volatile("s_wait_asynccnt 0" ::: "memory");
  __syncthreads();

  const int lane = threadIdx.x & 31;
  const int wave = threadIdx.x >> 5;
  const int gw   = blk * 8 + wave;

  const float outScale = (float)(DTd / 12.0);   // fold DT and x/12 into epilogue

  v2f Bones; Bones.x = 1.0f; Bones.y = 1.0f;

  for (int task = gw; task < N_ANG * NGRP; task += NBLK * 8) {
    const int ang  = task / NGRP;
    const int dgrp = task - ang * NGRP;
    const int m    = lane & 15;
    const int det  = dgrp * 16 + m;
    const int dc   = det < N_DET ? det : (N_DET - 1);   // dummy lanes compute, don't store

    // Per-task setup in f64 (matches reference's f64 coordinate generation)
    double theta = (ang + 0.5) * (3.14159265358979323846 / 285.0);
    double sn, cs;
    sincos(theta, &sn, &cs);
    double s = -RHO + (dc + 0.5) * SSTEP;

    const float Kr = (float)(cs * DTd * IDX);   // d(row idx)/d(sample)
    const float Kc = (float)(sn * DTd * IDX);   // d(col idx)/d(sample)

    // Each lane accumulates 2 of the 4 t-chunks (96 samples each):
    // lanes 0-15 -> chunks {0,1}, lanes 16-31 -> chunks {2,3}  == WMMA 16x4 f32 A layout
    float sums[2];
#pragma unroll
    for (int h = 0; h < 2; ++h) {
      const int chunk = ((lane >> 4) << 1) + h;
      double t00 = -RHO + (chunk * 96 + 0.5) * DTd;
      const float Br = (float)(((-s * sn + t00 * cs) + 20.0) * IDX - 0.5);
      const float Bc = (float)((( s * cs + t00 * sn) + 20.0) * IDX - 0.5);
      float acc = 0.0f;
#pragma unroll 4
      for (int i = 0; i < 96; ++i) {
        const float fi = (float)i;
        const float r = fmaf(fi, Kr, Br);
        const float c = fmaf(fi, Kc, Bc);
        if (r > -1.0f && r < 128.0f && c > -1.0f && c < 128.0f) {
          const float rf = floorf(r), cf = floorf(c);
          const float wr = r - rf,   wc = c - cf;
          const int ir = (int)rf, ic = (int)cf;
          const int ir0 = ir < 0 ? 0 : ir;
          const int ir1 = ir >= 127 ? 127 : ir + 1;
          const int ic0 = ic < 0 ? 0 : ic;
          const int ic1 = ic >= 127 ? 127 : ic + 1;
          const float w0r = (ir >= 0)  ? (1.0f - wr) : 0.0f;  // mode='constant', cval=0
          const float w1r = (ir < 127) ? wr : 0.0f;
          const float w0c = (ic >= 0)  ? (1.0f - wc) : 0.0f;
          const float w1c = (ic < 127) ? wc : 0.0f;
          const float* r0p = &simg[ir0 * PITCH];
          const float* r1p = &simg[ir1 * PITCH];
          acc = fmaf(w0r * w0c, r0p[ic0], acc);
          acc = fmaf(w0r * w1c, r0p[ic1], acc);
          acc = fmaf(w1r * w0c, r1p[ic0], acc);
          acc = fmaf(w1r * w1c, r1p[ic1], acc);
        }
      }
      sums[h] = acc;
    }

#if __has_builtin(__builtin_amdgcn_wmma_f32_16x16x4_f32)
    // Exact f32 cross-chunk reduction on the matrix unit:
    // D(16x16) = A(16 dets x 4 chunks) * ones(4x16); every column of D holds the ray sums.
    v2f A; A.x = sums[0]; A.y = sums[1];
    v8f C = {};
    v8f D = __builtin_amdgcn_wmma_f32_16x16x4_f32(
        false, A, false, Bones, (short)0, C, false, false);
    if ((lane & 15) == 0) {
      const int mbase = (lane >> 4) << 3;          // lane 0 -> dets m=0..7, lane 16 -> m=8..15
      float* orow = out + ((size_t)batch * N_ANG + ang) * N_DET;
#pragma unroll
      for (int rr = 0; rr < 8; ++rr) {
        const int d = dgrp * 16 + mbase + rr;
        if (d < N_DET) orow[d] = D[rr] * outScale;
      }
    }
#else
    // Fallback: shuffle reduction across the two half-wave chunk pairs
    float v = sums[0] + sums[1];
    v += __shfl_xor(v, 16, 32);
    if (lane < 16 && det < N_DET) {
      out[((size_t)batch * N_ANG + ang) * N_DET + det] = v * outScale;
    }
#endif
  }
}

extern "C" void kernel_launch(void* const* d_in, const int* in_sizes, int n_in,
                              void* d_out, int out_size, void* d_ws, size_t ws_size,
                              hipStream_t stream) {
  (void)in_sizes; (void)n_in; (void)d_ws; (void)ws_size; (void)out_size;
  const float* x = (const float*)d_in[0];
  float* out = (float*)d_out;
  dim3 grid(8 * NBLK), block(256);
  size_t shmem = (size_t)N_IMG * PITCH * sizeof(float);  // 66048 B
  hipLaunchKernelGGL(radon_fwd_kernel, grid, block, shmem, stream, x, out);
}